// TrajEmbedding_59966333387390
// MI455X (gfx1250) — compile-verified
//
#include <hip/hip_runtime.h>
#include <hip/hip_bf16.h>

typedef __attribute__((ext_vector_type(2))) float v2f;
typedef __attribute__((ext_vector_type(8))) float v8f;

#define N_NODES 50000
#define N_EDGES 800000
#define F_IN    128
#define D_EMB   128
#define BATCH   64
#define LMAX    2048

// Workspace layout (in floats):
//   [0, N_NODES)                     : deg, then overwritten in-place with dinv
//   [OFF_H,   OFF_H + N*D)           : h = x @ W
//   [OFF_ACC, OFF_ACC + N*D)         : edge-scatter accumulator -> node_emb (in place)
#define OFF_H   50176                      // N_NODES rounded up
#define OFF_ACC (OFF_H + N_NODES * D_EMB)

__global__ void zero_kernel(float* __restrict__ p, int n) {
    int i = blockIdx.x * blockDim.x + threadIdx.x;
    int stride = gridDim.x * blockDim.x;
    for (; i < n; i += stride) p[i] = 0.0f;
}

// deg[dst] += w  over real edges (self-loop weight 1 added in dinv_kernel)
__global__ void deg_kernel(const int* __restrict__ ei, const float* __restrict__ ew,
                           float* __restrict__ deg) {
    int e = blockIdx.x * blockDim.x + threadIdx.x;
    if (e < N_EDGES) unsafeAtomicAdd(&deg[ei[N_EDGES + e]], ew[e]);
}

// dinv = rsqrt(deg + 1)   (self-loop guarantees deg > 0, matching reference)
__global__ void dinv_kernel(float* __restrict__ deg) {
    int n = blockIdx.x * blockDim.x + threadIdx.x;
    if (n < N_NODES) deg[n] = rsqrtf(deg[n] + 1.0f);
}

// h = x @ W via V_WMMA_F32_16X16X4_F32.
// Block = 256 threads = 8 waves. Block handles 16 rows x 128 cols; wave w owns
// the 16x16 tile at columns [16w, 16w+16). K=128 swept in 32 steps of 4.
__global__ __launch_bounds__(256) void gemm_kernel(const float* __restrict__ x,
                                                   const float* __restrict__ W,
                                                   float* __restrict__ h) {
    __shared__ float xs[16][132];   // +4 pad: lanes 0-15 read a column (stride 132
                                    // -> 4 banks/row -> 16 distinct banks)
    const int r0 = blockIdx.x * 16;

    // Cooperatively stage the 16x128 A-panel (8 KB) into LDS.
    for (int t = threadIdx.x; t < 16 * 32; t += 256) {
        int m = t >> 5, c4 = t & 31;
        float4 v = ((const float4*)(x + (size_t)(r0 + m) * F_IN))[c4];
        xs[m][c4 * 4 + 0] = v.x; xs[m][c4 * 4 + 1] = v.y;
        xs[m][c4 * 4 + 2] = v.z; xs[m][c4 * 4 + 3] = v.w;
    }
    __syncthreads();

    const int wave = threadIdx.x >> 5;
    const int lane = threadIdx.x & 31;
    const int n0   = wave * 16;
    const int m    = lane & 15;     // doubles as N index for B and C/D
    const int half = lane >> 4;

    v8f c = {0.f, 0.f, 0.f, 0.f, 0.f, 0.f, 0.f, 0.f};
#pragma unroll
    for (int k0 = 0; k0 < F_IN; k0 += 4) {
        const int k = k0 + half * 2;
        v2f a, b;
        // A 16x4: lanes 0-15 -> K={k0,k0+1}, lanes 16-31 -> K={k0+2,k0+3}
        a.x = xs[m][k];
        a.y = xs[m][k + 1];
        // B 4x16: lane = N, same K split (W is row-major [K][N])
        b.x = W[(size_t)k * D_EMB + n0 + m];
        b.y = W[(size_t)(k + 1) * D_EMB + n0 + m];
        c = __builtin_amdgcn_wmma_f32_16x16x4_f32(false, a, false, b,
                                                  (short)0, c, false, false);
    }

    // C/D layout: VGPR v, lane half -> row M = v + 8*half, col N = lane&15
#pragma unroll
    for (int v = 0; v < 8; ++v) {
        int row = r0 + v + 8 * half;
        h[(size_t)row * D_EMB + n0 + m] = c[v];
    }
}

// One wave per edge: acc[dst] += norm * h[src], float4 per lane, f32 atomics.
__global__ __launch_bounds__(256) void scatter_kernel(const int* __restrict__ ei,
                                                      const float* __restrict__ ew,
                                                      const float* __restrict__ dinv,
                                                      const float* __restrict__ h,
                                                      float* __restrict__ acc) {
    int e = blockIdx.x * 8 + (threadIdx.x >> 5);
    if (e >= N_EDGES) return;
    const int lane = threadIdx.x & 31;
    const int src = ei[e];
    const int dst = ei[N_EDGES + e];
    const float norm = dinv[src] * ew[e] * dinv[dst];
    float4 hv = ((const float4*)(h + (size_t)src * D_EMB))[lane];
    float* a = acc + (size_t)dst * D_EMB + lane * 4;
    unsafeAtomicAdd(a + 0, hv.x * norm);
    unsafeAtomicAdd(a + 1, hv.y * norm);
    unsafeAtomicAdd(a + 2, hv.z * norm);
    unsafeAtomicAdd(a + 3, hv.w * norm);
}

// node_emb = relu(acc + h*dinv^2 (self-loop) + b), written over acc in place.
__global__ void finish_kernel(const float* __restrict__ h, const float* __restrict__ dinv,
                              const float* __restrict__ b, float* __restrict__ acc) {
    int i = blockIdx.x * blockDim.x + threadIdx.x;
    if (i >= N_NODES * D_EMB) return;
    int n = i >> 7, d = i & 127;
    float di = dinv[n];
    float v = acc[i] + h[i] * di * di + b[d];
    acc[i] = v > 0.f ? v : 0.f;
}

// One wave per (b,l): out = mask ? node_emb[traj[b,l]] : 0
__global__ __launch_bounds__(256) void gather_kernel(const int* __restrict__ traj,
                                                     const int* __restrict__ slen,
                                                     const float* __restrict__ emb,
                                                     float* __restrict__ out) {
    int pos = blockIdx.x * 8 + (threadIdx.x >> 5);
    if (pos >= BATCH * LMAX) return;
    const int lane = threadIdx.x & 31;
    const int bb = pos >> 11;      // / LMAX
    const int l  = pos & (LMAX - 1);
    float4 v = make_float4(0.f, 0.f, 0.f, 0.f);
    if (l < slen[bb]) {
        int node = traj[pos];
        v = ((const float4*)(emb + (size_t)node * D_EMB))[lane];
    }
    ((float4*)(out + (size_t)pos * D_EMB))[lane] = v;
}

// Second tuple output: seq_lengths (int32), copied raw into the output tail.
__global__ void seqlen_kernel(const int* __restrict__ slen, int* __restrict__ tail) {
    int i = threadIdx.x;
    if (i < BATCH) tail[i] = slen[i];
}

extern "C" void kernel_launch(void* const* d_in, const int* in_sizes, int n_in,
                              void* d_out, int out_size, void* d_ws, size_t ws_size,
                              hipStream_t stream) {
    const float* x    = (const float*)d_in[0];
    const int*   ei   = (const int*)  d_in[1];
    const float* ew   = (const float*)d_in[2];
    const float* W    = (const float*)d_in[3];
    const float* b    = (const float*)d_in[4];
    const int*   traj = (const int*)  d_in[5];
    const int*   slen = (const int*)  d_in[6];
    float* out = (float*)d_out;
    float* ws  = (float*)d_ws;

    float* deg = ws;            // becomes dinv in place
    float* h   = ws + OFF_H;
    float* acc = ws + OFF_ACC;

    zero_kernel<<<512, 256, 0, stream>>>(deg, N_NODES);
    zero_kernel<<<2048, 256, 0, stream>>>(acc, N_NODES * D_EMB);

    deg_kernel<<<(N_EDGES + 255) / 256, 256, 0, stream>>>(ei, ew, deg);
    dinv_kernel<<<(N_NODES + 255) / 256, 256, 0, stream>>>(deg);

    gemm_kernel<<<N_NODES / 16, 256, 0, stream>>>(x, W, h);          // 3125 blocks

    scatter_kernel<<<N_EDGES / 8, 256, 0, stream>>>(ei, ew, deg, h, acc);

    finish_kernel<<<(N_NODES * D_EMB + 255) / 256, 256, 0, stream>>>(h, deg, b, acc);

    gather_kernel<<<(BATCH * LMAX) / 8, 256, 0, stream>>>(traj, slen, acc, out);

    seqlen_kernel<<<1, 64, 0, stream>>>(slen, (int*)(out + (size_t)BATCH * LMAX * D_EMB));
}